// ClassAttention_64785286693553
// MI455X (gfx1250) — compile-verified
//
#include <hip/hip_runtime.h>

#define B 16
#define S 2049
#define D 1024
#define H 16
#define E 64
#define SCALE 0.125f
#define SCH 16
#define CS 129   // ceil(S/SCH)

typedef float v2f __attribute__((ext_vector_type(2)));
typedef float v4f __attribute__((ext_vector_type(4)));
typedef float v8f __attribute__((ext_vector_type(8)));

__device__ __forceinline__ float dot1024_v4(const float* __restrict__ a,
                                            const float* __restrict__ b) {
    const v4f* av = (const v4f*)a;
    const v4f* bv = (const v4f*)b;
    v4f acc = {0.f, 0.f, 0.f, 0.f};
#pragma unroll 8
    for (int i = 0; i < D / 4; ++i) {
        v4f x = av[i], w = bv[i];
        acc.x = fmaf(x.x, w.x, acc.x);
        acc.y = fmaf(x.y, w.y, acc.y);
        acc.z = fmaf(x.z, w.z, acc.z);
        acc.w = fmaf(x.w, w.w, acc.w);
    }
    return (acc.x + acc.y) + (acc.z + acc.w);
}

// ---- 1a: q_cls[b, r] = <x[b,0,:], w_qkv[r,:]>  (r in [0,D): the Q rows) ----
__global__ __launch_bounds__(256) void k_qcls(const float* __restrict__ x,
                                              const float* __restrict__ wqkv,
                                              float* __restrict__ qcls) {
    const int r = blockIdx.x * 256 + threadIdx.x;   // 0..D-1
    const int b = blockIdx.y;
    qcls[(size_t)b * D + r] =
        dot1024_v4(x + (size_t)b * S * D, wqkv + (size_t)r * D);
}

// ---- 1b: qk_vec[b,h,d] = SCALE * sum_e qcls[b,h*E+e] * w_k[h*E+e, d] ----
__global__ __launch_bounds__(256) void k_qkvec(const float* __restrict__ qcls,
                                               const float* __restrict__ wqkv,
                                               float* __restrict__ qk) {
    const int d = blockIdx.x * 256 + threadIdx.x;
    const int h = blockIdx.y, b = blockIdx.z;
    const float* q  = qcls + (size_t)b * D + h * E;
    const float* wk = wqkv + ((size_t)D + (size_t)h * E) * D + d;  // K rows start at D
    float acc = 0.f;
#pragma unroll 8
    for (int e = 0; e < E; ++e) acc = fmaf(q[e], wk[(size_t)e * D], acc);
    qk[((size_t)b * H + h) * D + d] = acc * SCALE;
}

// ---- 2: logits via WMMA f32 16x16x4 ----
// One wave owns a 16(s) x 16(h) tile. Virtual-K permutation: per b128 load,
// lane-half 0 carries physical d = k0..k0+3, half 1 carries d = k0+4..k0+7,
// identically for A (x rows) and B (qk rows), so the contraction pairing is
// consistent and each 16-byte load feeds two WMMA steps.
__global__ __launch_bounds__(256) void k_logits(const float* __restrict__ x,
                                                const float* __restrict__ qk,
                                                float* __restrict__ attnw) {
    const int wid  = threadIdx.x >> 5;
    const int lane = threadIdx.x & 31;
    const int b    = blockIdx.y;
    const int tile = blockIdx.x * 8 + wid;
    const int NT   = (S + 15) / 16;                 // 129 tiles per batch
    if (tile >= NT) return;
    const int s0   = tile * 16;
    const int l    = lane & 15;                     // row-in-tile for A; head for B/C
    const int half = lane >> 4;
    const int rowU = s0 + l;
    const int rowC = rowU < S ? rowU : (S - 1);

    const float* xp = x  + ((size_t)b * S + rowC) * D + 4 * half;
    const float* qp = qk + ((size_t)b * H + l)   * D + 4 * half;

    v8f c = {0.f, 0.f, 0.f, 0.f, 0.f, 0.f, 0.f, 0.f};
#pragma unroll 4
    for (int k0 = 0; k0 < D; k0 += 8) {
        v4f a = *(const v4f*)(xp + k0);
        v4f q = *(const v4f*)(qp + k0);
        v2f a0 = {a.x, a.y}, a1 = {a.z, a.w};
        v2f q0 = {q.x, q.y}, q1 = {q.z, q.w};
        c = __builtin_amdgcn_wmma_f32_16x16x4_f32(false, a0, false, q0,
                                                  (short)0, c, false, false);
        c = __builtin_amdgcn_wmma_f32_16x16x4_f32(false, a1, false, q1,
                                                  (short)0, c, false, false);
    }
    // C layout: vgpr r -> M = r + 8*half, N = l ; attnw layout [b][s][h]
    float* obase = attnw + ((size_t)b * S + s0 + 8 * half) * H + l;
    if (s0 + 16 <= S) {          // full tile: unconditional immediate-offset stores
#pragma unroll
        for (int r = 0; r < 8; ++r) obase[r * H] = c[r];
    } else {                     // tail tile only
#pragma unroll
        for (int r = 0; r < 8; ++r) {
            if (s0 + r + 8 * half < S) obase[r * H] = c[r];
        }
    }
}

// ---- 3: softmax over s for each (b,h); attnw layout [b][s][h] ----
__global__ __launch_bounds__(256) void k_softmax(float* __restrict__ attnw) {
    __shared__ float red[256];
    const int h = blockIdx.x, b = blockIdx.y, tid = threadIdx.x;
    float* base = attnw + (size_t)b * S * H + h;
    float m = -3.4e38f;
    for (int s = tid; s < S; s += 256) m = fmaxf(m, base[(size_t)s * H]);
    red[tid] = m; __syncthreads();
    for (int o = 128; o > 0; o >>= 1) {
        if (tid < o) red[tid] = fmaxf(red[tid], red[tid + o]);
        __syncthreads();
    }
    m = red[0]; __syncthreads();
    float sum = 0.f;
    for (int s = tid; s < S; s += 256) {
        float e = __expf(base[(size_t)s * H] - m);
        base[(size_t)s * H] = e;
        sum += e;
    }
    red[tid] = sum; __syncthreads();
    for (int o = 128; o > 0; o >>= 1) {
        if (tid < o) red[tid] += red[tid + o];
        __syncthreads();
    }
    const float inv = 1.f / red[0];
    for (int s = tid; s < S; s += 256) base[(size_t)s * H] *= inv;
}

// ---- 4: partial ctx + fused x -> out copy ----
// Block = (b, s-chunk); thread owns 4 d-columns (b128 loads/stores).
// Copy is unconditional: out[b,0,:] is overwritten later by k_proj (stream order).
__global__ __launch_bounds__(256) void k_ctx(const float* __restrict__ x,
                                             const float* __restrict__ attnw,
                                             float* __restrict__ ctxp,
                                             float* __restrict__ out) {
    __shared__ float wsh[64 * H];
    const int tid = threadIdx.x;
    const int d4  = tid * 4;
    const int b   = blockIdx.y;
    const int ck  = blockIdx.z;
    const int sBeg = ck * CS;
    const int sEnd = min(S, sBeg + CS);
    v4f acc[H];
#pragma unroll
    for (int h = 0; h < H; ++h) acc[h] = (v4f){0.f, 0.f, 0.f, 0.f};
    for (int s0 = sBeg; s0 < sEnd; s0 += 64) {
        const int n = min(64, sEnd - s0);
        const float* wbase = attnw + ((size_t)b * S + s0) * H;  // contiguous n*16
        for (int i = tid; i < 64 * H; i += 256)
            wsh[i] = (i < n * H) ? wbase[i] : 0.f;
        __syncthreads();
        for (int ss = 0; ss < n; ++ss) {
            const size_t off = ((size_t)b * S + s0 + ss) * D + d4;
            v4f xv = *(const v4f*)(x + off);
            *(v4f*)(out + off) = xv;                // fused copy (row 0 fixed later)
#pragma unroll
            for (int h = 0; h < H; ++h) {
                const float w = wsh[ss * H + h];
                acc[h].x = fmaf(w, xv.x, acc[h].x);
                acc[h].y = fmaf(w, xv.y, acc[h].y);
                acc[h].z = fmaf(w, xv.z, acc[h].z);
                acc[h].w = fmaf(w, xv.w, acc[h].w);
            }
        }
        __syncthreads();
    }
#pragma unroll
    for (int h = 0; h < H; ++h)
        *(v4f*)(ctxp + (((size_t)ck * B + b) * H + h) * D + d4) = acc[h];
}

// ---- 5: cls[b, h*E+e] = <sum_c ctxp[c][b,h,:], w_v[h*E+e,:]> ----
__global__ __launch_bounds__(64) void k_cls(const float* __restrict__ ctxp,
                                            const float* __restrict__ wqkv,
                                            float* __restrict__ cls) {
    __shared__ float row[D];
    const int h = blockIdx.x, b = blockIdx.y, e = threadIdx.x;
    for (int j = 0; j < D / 64; ++j) {
        const int d = e + j * 64;
        float s = 0.f;
#pragma unroll
        for (int c = 0; c < SCH; ++c)
            s += ctxp[(((size_t)c * B + b) * H + h) * D + d];
        row[d] = s;
    }
    __syncthreads();
    const float* wv = wqkv + ((size_t)(2 * D) + (size_t)h * E + e) * D;  // V rows at 2D
    cls[(size_t)b * D + (size_t)h * E + e] = dot1024_v4(row, wv);
}

// ---- 6: out[b,0,f] = <cls[b,:], w_proj[f,:]> + b_proj[f] ----
__global__ __launch_bounds__(256) void k_proj(const float* __restrict__ cls,
                                              const float* __restrict__ wproj,
                                              const float* __restrict__ bproj,
                                              float* __restrict__ out) {
    __shared__ float row[D];
    const int b = blockIdx.y;
    const int f = blockIdx.x * 256 + threadIdx.x;
    for (int j = threadIdx.x; j < D; j += 256) row[j] = cls[(size_t)b * D + j];
    __syncthreads();
    out[(size_t)b * S * D + f] = bproj[f] + dot1024_v4(row, wproj + (size_t)f * D);
}

extern "C" void kernel_launch(void* const* d_in, const int* in_sizes, int n_in,
                              void* d_out, int out_size, void* d_ws, size_t ws_size,
                              hipStream_t stream) {
    (void)in_sizes; (void)n_in; (void)out_size; (void)ws_size;
    const float* x     = (const float*)d_in[0];
    const float* wqkv  = (const float*)d_in[1];
    const float* wproj = (const float*)d_in[2];
    const float* bproj = (const float*)d_in[3];
    float* out = (float*)d_out;

    float* ws    = (float*)d_ws;
    float* qcls  = ws;                                    // B*D
    float* qk    = qcls  + (size_t)B * D;                 // B*H*D
    float* attnw = qk    + (size_t)B * H * D;             // B*S*H
    float* ctxp  = attnw + (size_t)B * S * H;             // SCH*B*H*D
    float* cls   = ctxp  + (size_t)SCH * B * H * D;       // B*D

    k_qcls   <<<dim3(D / 256, B),      256, 0, stream>>>(x, wqkv, qcls);
    k_qkvec  <<<dim3(D / 256, H, B),   256, 0, stream>>>(qcls, wqkv, qk);
    k_logits <<<dim3(17, B),           256, 0, stream>>>(x, qk, attnw);
    k_softmax<<<dim3(H, B),            256, 0, stream>>>(attnw);
    k_ctx    <<<dim3(1, B, SCH),       256, 0, stream>>>(x, attnw, ctxp, out);
    k_cls    <<<dim3(H, B),            64,  0, stream>>>(ctxp, wqkv, cls);
    k_proj   <<<dim3(D / 256, B),      256, 0, stream>>>(cls, wproj, bproj, out);
}